// LightfieldViewerModel_84189948936747
// MI455X (gfx1250) — compile-verified
//
#include <hip/hip_runtime.h>

// Lightfield quadrilinear interpolation for MI455X (gfx1250, wave32).
// Gather-bound kernel: ~2 GB of random corner gathers into a 511 MB table,
// bounded by HBM (23.3 TB/s) at ~60-90 us. No legal WMMA mapping exists
// (per-row corner matrices, no shared B operand), so the optimization targets
// are MLP (48 outstanding loads/thread), wave32 occupancy, and L2 reservation
// for the gather table via non-temporal hints on streaming data.

#define LNU 17
#define LNV 17
#define LNX 384
#define LNY 384
#define LC  3

struct AxisP { float i0f; float w; int oob; };

__device__ __forceinline__ AxisP axis_params(float q, float g0, float g1,
                                             float glast, int n) {
    float step = g1 - g0;
    float t = (q - g0) / step;                 // match reference: true divide
    float i0f = floorf(t);
    i0f = fminf(fmaxf(i0f, 0.0f), (float)(n - 2));
    AxisP r;
    r.i0f = i0f;
    r.w   = t - i0f;
    r.oob = (q < g0) | (q > glast);
    return r;
}

__global__ __launch_bounds__(256) void lf_interp_kernel(
    const float* __restrict__ lf,
    const float* __restrict__ imXY,
    const float* __restrict__ imUV,
    const float* __restrict__ gu, const float* __restrict__ gv,
    const float* __restrict__ gx, const float* __restrict__ gy,
    const int*   __restrict__ zsep,
    float*       __restrict__ out,
    int npix)
{
    int pix = blockIdx.x * blockDim.x + threadIdx.x;
    if (pix >= npix) return;

    // Streaming, read-once coordinates: non-temporal so they don't evict
    // lightfield lines from the 192MB L2.
    const float* pxy = imXY + (size_t)pix * 3;
    const float* puv = imUV + (size_t)pix * 3;
    float xy0 = __builtin_nontemporal_load(pxy + 0);
    float xy1 = __builtin_nontemporal_load(pxy + 1);
    float uv0 = __builtin_nontemporal_load(puv + 0);
    float uv1 = __builtin_nontemporal_load(puv + 1);

    float zs = (float)zsep[0];
    float d0 = (uv0 - xy0) / zs;   // dirs.x
    float d1 = (uv1 - xy1) / zs;   // dirs.y

    // query coords: ix = (-dirs.y, dirs.x, -xy.y, xy.x) vs grids (u,v,x,y)
    float qu = -d1;
    float qv =  d0;
    float qx = -xy1;
    float qy =  xy0;

    AxisP au = axis_params(qu, gu[0], gu[1], gu[LNU - 1], LNU);
    AxisP av = axis_params(qv, gv[0], gv[1], gv[LNV - 1], LNV);
    AxisP ax = axis_params(qx, gx[0], gx[1], gx[LNX - 1], LNX);
    AxisP ay = axis_params(qy, gy[0], gy[1], gy[LNY - 1], LNY);

    // element strides; full table < 2^32 bytes -> 32-bit offsets (GVS saddr form)
    const unsigned sY = LC;                    // 3
    const unsigned sX = LNY * sY;              // 1152
    const unsigned sV = LNX * sX;              // 442368
    const unsigned sU = LNV * sV;              // 7520256

    unsigned iu = (unsigned)(int)au.i0f;
    unsigned iv = (unsigned)(int)av.i0f;
    unsigned ix = (unsigned)(int)ax.i0f;
    unsigned iy = (unsigned)(int)ay.i0f;
    unsigned base = iu * sU + iv * sV + ix * sX + iy * sY;

    float wu1 = au.w, wu0 = 1.0f - wu1;
    float wv1 = av.w, wv0 = 1.0f - wv1;
    float wx1 = ax.w, wx0 = 1.0f - wx1;
    float wy1 = ay.w, wy0 = 1.0f - wy1;

    float acc0 = 0.0f, acc1 = 0.0f, acc2 = 0.0f;

    // 8 corner groups; dy=0/1 are 6 contiguous floats -> one cacheline group.
    // Full unroll => all 48 loads in flight before first use (deep MLP).
#pragma unroll
    for (int du = 0; du < 2; ++du) {
#pragma unroll
        for (int dv = 0; dv < 2; ++dv) {
#pragma unroll
            for (int dx = 0; dx < 2; ++dx) {
                unsigned off = base + (unsigned)du * sU
                                    + (unsigned)dv * sV
                                    + (unsigned)dx * sX;
                const float* p = lf + (size_t)off;
                float c00 = p[0], c01 = p[1], c02 = p[2];   // dy = 0
                float c10 = p[3], c11 = p[4], c12 = p[5];   // dy = 1
                float wuv = (du ? wu1 : wu0) * (dv ? wv1 : wv0) * (dx ? wx1 : wx0);
                float w0 = wuv * wy0;
                float w1 = wuv * wy1;
                acc0 = fmaf(w0, c00, fmaf(w1, c10, acc0));
                acc1 = fmaf(w0, c01, fmaf(w1, c11, acc1));
                acc2 = fmaf(w0, c02, fmaf(w1, c12, acc2));
            }
        }
    }

    int oob = au.oob | av.oob | ax.oob | ay.oob;
    if (oob) { acc0 = 0.0f; acc1 = 0.0f; acc2 = 0.0f; }

    acc0 = fminf(fmaxf(acc0, 0.0f), 1.0f);
    acc1 = fminf(fmaxf(acc1, 0.0f), 1.0f);
    acc2 = fminf(fmaxf(acc2, 0.0f), 1.0f);

    // Write-once output: non-temporal store keeps L2 for the gather table.
    float* po = out + (size_t)pix * 3;
    __builtin_nontemporal_store(acc0, po + 0);
    __builtin_nontemporal_store(acc1, po + 1);
    __builtin_nontemporal_store(acc2, po + 2);
}

extern "C" void kernel_launch(void* const* d_in, const int* in_sizes, int n_in,
                              void* d_out, int out_size, void* d_ws, size_t ws_size,
                              hipStream_t stream) {
    const float* lf   = (const float*)d_in[0];   // lightfield [17,17,384,384,3]
    const float* imXY = (const float*)d_in[1];   // [8,512,512,3]
    const float* imUV = (const float*)d_in[2];   // [8,512,512,3]
    const float* gu   = (const float*)d_in[3];   // [17]
    const float* gv   = (const float*)d_in[4];   // [17]
    const float* gx   = (const float*)d_in[5];   // [384]
    const float* gy   = (const float*)d_in[6];   // [384]
    const int*   zsep = (const int*)d_in[7];     // scalar

    float* out = (float*)d_out;
    int npix = in_sizes[1] / 3;                  // 8*512*512 = 2,097,152

    dim3 block(256);                             // 8 wave32 waves per block
    dim3 grid((npix + 255) / 256);
    hipLaunchKernelGGL(lf_interp_kernel, grid, block, 0, stream,
                       lf, imXY, imUV, gu, gv, gx, gy, zsep, out, npix);
}